// Complex_Loss_25391846654425
// MI455X (gfx1250) — compile-verified
//
#include <hip/hip_runtime.h>
#include <hip/hip_bf16.h>

typedef __attribute__((ext_vector_type(2))) float v2f;
typedef __attribute__((ext_vector_type(8))) float v8f;

#define D_HALF 4096
#define ROW_STRIDE 8192
#define THREADS 256

// ---------------------------------------------------------------------------
// Kernel 1: one workgroup (8 wave32) per row. Streams 4 contiguous fp32
// vectors per row (out_r, out_i, gt_r, gt_i) with float4 (global_load_b128)
// loads, computes sum(sq), max(sq), argmax(gt_r), argmax(out_r), applies the
// mask, and writes per-row (masked_sum, masked_count) partials to scratch.
// Purely HBM-bandwidth-bound: 1.07 GB @ 23.3 TB/s => ~46 us floor.
// ---------------------------------------------------------------------------
__global__ __launch_bounds__(THREADS) void row_reduce_kernel(
    const float* __restrict__ out,
    const float* __restrict__ g_t,
    const int*   __restrict__ epoch_p,
    float* __restrict__ ws_sum,
    float* __restrict__ ws_cnt)
{
    const int row = blockIdx.x;
    const int t   = threadIdx.x;

    const float* out_r = out + (size_t)row * ROW_STRIDE;
    const float* out_i = out_r + D_HALF;
    const float* gt_r  = g_t + (size_t)row * ROW_STRIDE;
    const float* gt_i  = gt_r + D_HALF;

    float sum   = 0.0f;
    float maxsq = -__builtin_inff();
    float gmax  = -__builtin_inff(); int gidx = 0;
    float omax  = -__builtin_inff(); int oidx = 0;

    // 4096 floats / 256 threads = 16 per thread, as 4 chunks of float4.
    // Within a thread j increases monotonically -> strict '>' keeps the
    // FIRST occurrence of the max (matches jnp.argmax tie semantics).
#pragma unroll
    for (int c = 0; c < 4; ++c) {
        const int j = c * 1024 + t * 4;
        const float4 orv = *(const float4*)(out_r + j);
        const float4 oiv = *(const float4*)(out_i + j);
        const float4 grv = *(const float4*)(gt_r  + j);
        const float4 giv = *(const float4*)(gt_i  + j);
        const float* orp = (const float*)&orv;
        const float* oip = (const float*)&oiv;
        const float* grp = (const float*)&grv;
        const float* gip = (const float*)&giv;
#pragma unroll
        for (int k = 0; k < 4; ++k) {
            const float er = grp[k] - orp[k];
            const float ei = gip[k] - oip[k];
            const float sq = fmaf(er, er, ei * ei);
            sum += sq;
            maxsq = fmaxf(maxsq, sq);
            if (grp[k] > gmax) { gmax = grp[k]; gidx = j + k; }
            if (orp[k] > omax) { omax = orp[k]; oidx = j + k; }
        }
    }

    __shared__ float s_sum[THREADS];
    __shared__ float s_max[THREADS];
    __shared__ float s_gv[THREADS];  __shared__ int s_gi[THREADS];
    __shared__ float s_ov[THREADS];  __shared__ int s_oi[THREADS];

    s_sum[t] = sum;  s_max[t] = maxsq;
    s_gv[t] = gmax;  s_gi[t] = gidx;
    s_ov[t] = omax;  s_oi[t] = oidx;
    __syncthreads();

    for (int off = THREADS / 2; off > 0; off >>= 1) {
        if (t < off) {
            s_sum[t] += s_sum[t + off];
            s_max[t]  = fmaxf(s_max[t], s_max[t + off]);
            // argmax combine: larger value wins; tie -> smaller index (first occurrence)
            {
                const float v2 = s_gv[t + off]; const int i2 = s_gi[t + off];
                if (v2 > s_gv[t] || (v2 == s_gv[t] && i2 < s_gi[t])) { s_gv[t] = v2; s_gi[t] = i2; }
            }
            {
                const float v2 = s_ov[t + off]; const int i2 = s_oi[t + off];
                if (v2 > s_ov[t] || (v2 == s_ov[t] && i2 < s_oi[t])) { s_ov[t] = v2; s_oi[t] = i2; }
            }
        }
        __syncthreads();
    }

    if (t == 0) {
        const int epoch = epoch_p[0];
        // 0.5 * exp(-0.2) evaluated in double then truncated to f32 (matches numpy)
        const float e_thresh = (epoch % 10 == 0) ? 0.40936537653899087f : 0.5f;
        const bool correct = (s_gi[0] == s_oi[0]);
        const bool masked  = correct && (sqrtf(s_max[0]) < e_thresh);
        ws_sum[row] = masked ? 0.0f : s_sum[0];
        ws_cnt[row] = masked ? 1.0f : (float)D_HALF;
    }
}

// ---------------------------------------------------------------------------
// Kernel 2: single wave32 final reduction over the per-row partials using
// V_WMMA_F32_16X16X4_F32 with B = ones: D[m][n] = sum_k A[m][k] + C[m][n].
// Each WMMA folds 64 fp32 partials into the accumulator; 256 iterations
// cover 16384 rows. Deterministic fixed-order reduction. Touches 128 KB —
// negligible vs. the 1 GB streaming pass.
// ---------------------------------------------------------------------------
__global__ __launch_bounds__(32) void final_reduce_kernel(
    const float* __restrict__ ws_sum,
    const float* __restrict__ ws_cnt,
    float* __restrict__ d_out,
    int nrows)
{
    const int lane = threadIdx.x;   // full wave32, EXEC all ones for WMMA
    const v2f b_ones = {1.0f, 1.0f};
    v8f csum = {};
    v8f ccnt = {};

    const int iters = nrows / 64;   // 64 fp32 elements folded per WMMA
    for (int it = 0; it < iters; ++it) {
        const int base = it * 64 + lane * 2;
        const v2f a0 = *(const v2f*)(ws_sum + base);
        const v2f a1 = *(const v2f*)(ws_cnt + base);
        csum = __builtin_amdgcn_wmma_f32_16x16x4_f32(
            false, a0, false, b_ones, (short)0, csum, false, false);
        ccnt = __builtin_amdgcn_wmma_f32_16x16x4_f32(
            false, a1, false, b_ones, (short)0, ccnt, false, false);
    }

    // Column N=0 of D lives in lane 0 (M=0..7 in VGPRs 0..7) and lane 16
    // (M=8..15). Sum the 8 accumulator VGPRs per lane, then pull lanes 0+16.
    float tsum = 0.0f, tcnt = 0.0f;
#pragma unroll
    for (int v = 0; v < 8; ++v) { tsum += csum[v]; tcnt += ccnt[v]; }

    const float total_sum = __shfl(tsum, 0) + __shfl(tsum, 16);
    const float total_cnt = __shfl(tcnt, 0) + __shfl(tcnt, 16);

    if (lane == 0) d_out[0] = total_sum / (1.0f + total_cnt);
}

// ---------------------------------------------------------------------------
extern "C" void kernel_launch(void* const* d_in, const int* in_sizes, int n_in,
                              void* d_out, int out_size, void* d_ws, size_t ws_size,
                              hipStream_t stream) {
    const float* out_m  = (const float*)d_in[0];
    const float* g_t    = (const float*)d_in[1];
    const int*   epoch  = (const int*)d_in[2];

    const int nrows = in_sizes[0] / ROW_STRIDE;   // 16384

    float* ws_sum = (float*)d_ws;
    float* ws_cnt = ws_sum + nrows;

    row_reduce_kernel<<<nrows, THREADS, 0, stream>>>(out_m, g_t, epoch, ws_sum, ws_cnt);
    final_reduce_kernel<<<1, 32, 0, stream>>>(ws_sum, ws_cnt, (float*)d_out, nrows);
}